// GroupWiseCorrelation_91199335563538
// MI455X (gfx1250) — compile-verified
//
#include <hip/hip_runtime.h>

typedef __attribute__((ext_vector_type(2))) float v2f;
typedef __attribute__((ext_vector_type(8))) float v8f;

namespace {
constexpr int kB   = 4;
constexpr int kC   = 256;
constexpr int kG   = 4;
constexpr int kCpg = kC / kG;          // 64 channels per group
constexpr int kH   = 96;
constexpr int kW   = 160;
constexpr int kP   = 9;
constexpr int kPad = 4;                // (P/2)*DIL
constexpr int kWT  = kW / 16;          // 10 w-tiles
constexpr int kCS  = kH * kW;          // channel stride (elements)
constexpr float kNeg = 0.1f;
}

__global__ __launch_bounds__(32)
void corr_wmma_f32_kernel(const float* __restrict__ x1,
                          const float* __restrict__ x2,
                          float* __restrict__ out) {
  // Two 16x16 f32 result tiles staged in LDS for diagonal-band extraction.
  __shared__ float lds[512];

  const int bid = blockIdx.x;
  const int wt  = bid % kWT;
  const int h   = (bid / kWT) % kH;
  const int g   = (bid / (kWT * kH)) % kG;
  const int b   = bid / (kWT * kH * kG);
  const int w0  = wt * 16;

  const int lane = threadIdx.x & 31;
  const int half = lane >> 4;          // 0: lanes 0-15, 1: lanes 16-31
  const int l16  = lane & 15;

  const float* aBase = x1 + (unsigned)((b * kC + g * kCpg) * kCS);
  const float* bBase = x2 + (unsigned)((b * kC + g * kCpg) * kCS);

  // ---- A fragments (16x64 tile of x1 at row h), reused for all 9 pi ----
  // A layout (f32 16x16x4): lanes 0-15 hold K=0,1; lanes 16-31 hold K=2,3.
  // One per-lane base address; per-k channel offsets fold into imm offsets.
  v2f A[16];
  {
    const float* ap = aBase + (unsigned)(h * kW + w0 + l16);
    #pragma unroll
    for (int k = 0; k < 16; ++k) {
      const unsigned c0 = 4u * k + 2u * half;
      A[k].x = ap[c0 * (unsigned)kCS];
      A[k].y = ap[(c0 + 1u) * (unsigned)kCS];
    }
  }

  const bool interior = (wt > 0) && (wt < kWT - 1);  // wave-uniform
  const int  wA = w0 - kPad + l16;                   // tile0 column

  const int outBase  = (((b * kG + g) * kP) * kP) * (kH * kW) + h * kW + w0;
  const int piStride = kP * kH * kW;
  const int pjStride = kH * kW;

  for (int pi = 0; pi < kP; ++pi) {
    const int hh = h + pi - kPad;
    const int ob = outBase + pi * piStride;

    if (hh >= 0 && hh < kH) {          // wave-uniform branch
      // ---- B fragments: 64 x 32 tile of x2 at row hh, cols w0-4 .. w0+27 ----
      const float* bp = bBase + (unsigned)(hh * kW);  // uniform base -> saddr
      v2f B0[16], B1[16];

      if (interior) {
        // Fast path: all columns in range, no selects. Unsigned 32-bit
        // offsets -> global_load saddr + voffset + imm (B1 = same voff +64B).
        const unsigned base = (unsigned)wA;
        #pragma unroll
        for (int k = 0; k < 16; ++k) {
          const unsigned c0 = 4u * k + 2u * half;
          const unsigned r0 = c0 * (unsigned)kCS;
          const unsigned r1 = (c0 + 1u) * (unsigned)kCS;
          B0[k].x = bp[r0 + base];
          B0[k].y = bp[r1 + base];
          B1[k].x = bp[r0 + base + 16u];
          B1[k].y = bp[r1 + base + 16u];
        }
      } else {
        // Boundary w-tile: clamp addresses, select zero for padded columns.
        const int  wB  = wA + 16;
        const bool okA = (wA >= 0);
        const bool okB = (wB < kW);
        const unsigned wAc = (unsigned)(okA ? wA : 0);
        const unsigned wBc = (unsigned)(okB ? wB : (kW - 1));
        #pragma unroll
        for (int k = 0; k < 16; ++k) {
          const unsigned c0 = 4u * k + 2u * half;
          const unsigned r0 = c0 * (unsigned)kCS;
          const unsigned r1 = (c0 + 1u) * (unsigned)kCS;
          const float a0 = bp[r0 + wAc];
          const float a1 = bp[r1 + wAc];
          const float b0 = bp[r0 + wBc];
          const float b1 = bp[r1 + wBc];
          B0[k].x = okA ? a0 : 0.0f;
          B0[k].y = okA ? a1 : 0.0f;
          B1[k].x = okB ? b0 : 0.0f;
          B1[k].y = okB ? b1 : 0.0f;
        }
      }

      // ---- K=64 accumulation via 16 x (16x16x4 f32) WMMA per N-tile ----
      v8f acc0 = {};
      v8f acc1 = {};
      #pragma unroll
      for (int k = 0; k < 16; ++k) {
        acc0 = __builtin_amdgcn_wmma_f32_16x16x4_f32(
            false, A[k], false, B0[k], (short)0, acc0, false, false);
        acc1 = __builtin_amdgcn_wmma_f32_16x16x4_f32(
            false, A[k], false, B1[k], (short)0, acc1, false, false);
      }

      // ---- Stage D tiles to LDS (VGPR v -> M = v + 8*half, N = l16) ----
      #pragma unroll
      for (int v = 0; v < 8; ++v) {
        const int m = v + 8 * half;
        lds[m * 16 + l16]       = acc0[v];
        lds[256 + m * 16 + l16] = acc1[v];
      }
      __syncthreads();

      // ---- Extract band out[m,pj] = D[m, m+pj], leaky ReLU, store ----
      #pragma unroll
      for (int jj = 0; jj < 5; ++jj) {
        const int pj = 2 * jj + half;  // lanes 0-15: even pj, 16-31: odd pj
        if (pj < kP) {
          const int m = l16;
          const int n = m + pj;
          float val = (n < 16) ? lds[m * 16 + n] : lds[256 + m * 16 + (n - 16)];
          val = (val >= 0.0f) ? val : kNeg * val;
          __builtin_nontemporal_store(val, &out[ob + pj * pjStride + m]);
        }
      }
      __syncthreads();  // protect LDS reuse across pi iterations
    } else {
      // Row shift falls entirely in zero padding: output is exactly zero.
      #pragma unroll
      for (int jj = 0; jj < 5; ++jj) {
        const int pj = 2 * jj + half;
        if (pj < kP) {
          __builtin_nontemporal_store(0.0f, &out[ob + pj * pjStride + l16]);
        }
      }
    }
  }
}

extern "C" void kernel_launch(void* const* d_in, const int* in_sizes, int n_in,
                              void* d_out, int out_size, void* d_ws, size_t ws_size,
                              hipStream_t stream) {
  const float* x1 = (const float*)d_in[0];
  const float* x2 = (const float*)d_in[1];
  float* out = (float*)d_out;
  (void)in_sizes; (void)n_in; (void)out_size; (void)d_ws; (void)ws_size;

  const int grid = kB * kG * kH * kWT;  // 4*4*96*10 = 15360 single-wave blocks
  corr_wmma_f32_kernel<<<grid, 32, 0, stream>>>(x1, x2, out);
}